// AstAttGRUModelUML_65274912964679
// MI455X (gfx1250) — compile-verified
//
#include <hip/hip_runtime.h>
#include <math.h>

#define BATCH    64
#define TLEN     30
#define STEPS    29
#define VOCAB    30000
#define HID      256
#define EMBD     256
#define CODELEN  200
#define SBTLEN   500

typedef __attribute__((ext_vector_type(2))) float v2f;
typedef __attribute__((ext_vector_type(8))) float v8f;

// ---------------------------------------------------------------------------
// Setup: class query/value (time-invariant), h <- init_hidden, tok <- ms[:,0]
// ---------------------------------------------------------------------------
__global__ void setup_kernel(const float* __restrict__ class_enc,
                             const float* __restrict__ class_feat,
                             const float* __restrict__ init_hidden,
                             const int*   __restrict__ method_summary,
                             const float* __restrict__ Wq_uml, const float* __restrict__ bq_uml,
                             const float* __restrict__ Wq_name, const float* __restrict__ bq_name,
                             const float* __restrict__ Wv_uml, const float* __restrict__ bv_uml,
                             const float* __restrict__ Wv_name, const float* __restrict__ bv_name,
                             float* __restrict__ h_ws, float* __restrict__ cq_ws,
                             float* __restrict__ cv_ws, int* __restrict__ tok_ws)
{
    const int b = blockIdx.x;
    const int j = threadIdx.x;
    __shared__ float ce[HID];
    __shared__ float cf[HID];
    ce[j] = class_enc[b * HID + j];
    cf[j] = class_feat[b * HID + j];
    __syncthreads();

    float q0 = bq_uml[j], q1 = bq_name[j], v0 = bv_uml[j], v1 = bv_name[j];
    const float* wqu = Wq_uml + (size_t)j * HID;
    const float* wqn = Wq_name + (size_t)j * HID;
    const float* wvu = Wv_uml + (size_t)j * HID;
    const float* wvn = Wv_name + (size_t)j * HID;
    for (int k = 0; k < HID; ++k) {
        const float c = ce[k], f = cf[k];
        q0 += c * wqu[k];
        q1 += f * wqn[k];
        v0 += c * wvu[k];
        v1 += f * wvn[k];
    }
    cq_ws[b * 2 * HID + j]       = q0;
    cq_ws[b * 2 * HID + HID + j] = q1;
    cv_ws[b * 2 * HID + j]       = v0;
    cv_ws[b * 2 * HID + HID + j] = v1;
    h_ws[b * HID + j] = init_hidden[b * HID + j];
    if (j == 0) tok_ws[b] = method_summary[b * TLEN];
}

// ---------------------------------------------------------------------------
// Per-step: GRU cell + code/sbt softmax attention + class attention + fc/ReLU
// One block (256 threads) per batch row.
// fc layout: [cls_c | ast_c(sbt) | txt_c(code) | h]
// ---------------------------------------------------------------------------
__global__ void gru_att_kernel(const int*   __restrict__ tok_ws,
                               const float* __restrict__ emb,
                               const float* __restrict__ W_ih, const float* __restrict__ W_hh,
                               const float* __restrict__ b_ih, const float* __restrict__ b_hh,
                               const float* __restrict__ code_enc,
                               const float* __restrict__ sbt_enc,
                               const float* __restrict__ cq_ws, const float* __restrict__ cv_ws,
                               const float* __restrict__ W1, const float* __restrict__ b1,
                               float* __restrict__ h_ws, float* __restrict__ hidden2_ws)
{
    const int b = blockIdx.x;
    const int j = threadIdx.x;
    __shared__ float xs[HID];
    __shared__ float hp[HID];
    __shared__ float hn[HID];
    __shared__ float fc[4 * HID];
    __shared__ float sc[SBTLEN];
    __shared__ float red[HID];

    const int tok = tok_ws[b];
    xs[j] = emb[(size_t)tok * EMBD + j];
    hp[j] = h_ws[b * HID + j];
    __syncthreads();

    // ---- GRU cell (gate order r,z,n as in torch GRUCell) ----
    float gr = b_ih[j], gz = b_ih[HID + j], gn = b_ih[2 * HID + j];
    float hr = b_hh[j], hz = b_hh[HID + j], hnv = b_hh[2 * HID + j];
    const float* wr = W_ih + (size_t)j * HID;
    const float* wz = W_ih + (size_t)(HID + j) * HID;
    const float* wn = W_ih + (size_t)(2 * HID + j) * HID;
    const float* ur = W_hh + (size_t)j * HID;
    const float* uz = W_hh + (size_t)(HID + j) * HID;
    const float* un = W_hh + (size_t)(2 * HID + j) * HID;
    for (int k = 0; k < HID; ++k) {
        const float xk = xs[k], hk = hp[k];
        gr += wr[k] * xk; gz += wz[k] * xk; gn += wn[k] * xk;
        hr += ur[k] * hk; hz += uz[k] * hk; hnv += un[k] * hk;
    }
    const float r = 1.0f / (1.0f + __expf(-(gr + hr)));
    const float z = 1.0f / (1.0f + __expf(-(gz + hz)));
    const float n = tanhf(gn + r * hnv);
    const float hval = (1.0f - z) * n + z * hp[j];
    hn[j] = hval;
    fc[3 * HID + j] = hval;
    h_ws[b * HID + j] = hval;
    __syncthreads();

    // ---- code (txt) attention: softmax over 200 keys ----
    {
        const float* cbase = code_enc + (size_t)b * CODELEN * HID;
        for (int i = j; i < CODELEN; i += HID) {
            const float* row = cbase + (size_t)i * HID;
            float s = 0.0f;
            for (int k = 0; k < HID; ++k) s += hn[k] * row[k];
            sc[i] = s;
        }
        __syncthreads();
        float lm = -3.4e38f;
        for (int i = j; i < CODELEN; i += HID) lm = fmaxf(lm, sc[i]);
        red[j] = lm; __syncthreads();
        for (int off = 128; off > 0; off >>= 1) { if (j < off) red[j] = fmaxf(red[j], red[j + off]); __syncthreads(); }
        const float mx = red[0]; __syncthreads();
        float ls = 0.0f;
        for (int i = j; i < CODELEN; i += HID) { const float e = __expf(sc[i] - mx); sc[i] = e; ls += e; }
        red[j] = ls; __syncthreads();
        for (int off = 128; off > 0; off >>= 1) { if (j < off) red[j] += red[j + off]; __syncthreads(); }
        const float inv = 1.0f / red[0]; __syncthreads();
        float ctx = 0.0f;
        for (int i = 0; i < CODELEN; ++i) ctx += sc[i] * cbase[(size_t)i * HID + j];
        fc[2 * HID + j] = ctx * inv;
        __syncthreads();
    }

    // ---- sbt (ast) attention: softmax over 500 keys ----
    {
        const float* sbase = sbt_enc + (size_t)b * SBTLEN * HID;
        for (int i = j; i < SBTLEN; i += HID) {
            const float* row = sbase + (size_t)i * HID;
            float s = 0.0f;
            for (int k = 0; k < HID; ++k) s += hn[k] * row[k];
            sc[i] = s;
        }
        __syncthreads();
        float lm = -3.4e38f;
        for (int i = j; i < SBTLEN; i += HID) lm = fmaxf(lm, sc[i]);
        red[j] = lm; __syncthreads();
        for (int off = 128; off > 0; off >>= 1) { if (j < off) red[j] = fmaxf(red[j], red[j + off]); __syncthreads(); }
        const float mx = red[0]; __syncthreads();
        float ls = 0.0f;
        for (int i = j; i < SBTLEN; i += HID) { const float e = __expf(sc[i] - mx); sc[i] = e; ls += e; }
        red[j] = ls; __syncthreads();
        for (int off = 128; off > 0; off >>= 1) { if (j < off) red[j] += red[j + off]; __syncthreads(); }
        const float inv = 1.0f / red[0]; __syncthreads();
        float ctx = 0.0f;
        for (int i = 0; i < SBTLEN; ++i) ctx += sc[i] * sbase[(size_t)i * HID + j];
        fc[HID + j] = ctx * inv;
        __syncthreads();
    }

    // ---- class attention (NO softmax — faithful to reference typo) ----
    {
        const float* cq = cq_ws + b * 2 * HID;
        const float* cv = cv_ws + b * 2 * HID;
        red[j] = hn[j] * cq[j]; __syncthreads();
        for (int off = 128; off > 0; off >>= 1) { if (j < off) red[j] += red[j + off]; __syncthreads(); }
        const float a0 = red[0]; __syncthreads();
        red[j] = hn[j] * cq[HID + j]; __syncthreads();
        for (int off = 128; off > 0; off >>= 1) { if (j < off) red[j] += red[j + off]; __syncthreads(); }
        const float a1 = red[0]; __syncthreads();
        fc[j] = a0 * cv[j] + a1 * cv[HID + j];
        __syncthreads();
    }

    // ---- hidden2 = relu(fc @ W1.T + b1) ----
    float acc = b1[j];
    const float* w1r = W1 + (size_t)j * (4 * HID);
    for (int m = 0; m < 4 * HID; ++m) acc += w1r[m] * fc[m];
    hidden2_ws[b * HID + j] = fmaxf(acc, 0.0f);
}

// ---------------------------------------------------------------------------
// logits[64,30000] = hidden2 @ W2^T + b2 via V_WMMA_F32_16X16X4_F32.
// 625 blocks x 96 threads (3 waves). Block stages 48 W2 rows (48x256 f32,
// LDS row stride 260 to avoid bank conflicts); each wave owns a 16-col tile
// and computes all M=64 rows (4 accumulators), reusing its B fragment 4x.
// Writes directly to d_out in [B, STEPS, V] layout with bias added.
// ---------------------------------------------------------------------------
__global__ void logits_wmma_kernel(const float* __restrict__ hidden2_ws,
                                   const float* __restrict__ W2,
                                   const float* __restrict__ b2,
                                   float* __restrict__ out, int step)
{
    __shared__ float bt[48 * 260];
    const int tid  = threadIdx.x;
    const int lane = tid & 31;
    const int wave = tid >> 5;
    const int n0b  = blockIdx.x * 48;            // 625 * 48 == 30000 exactly

    // stage W2[n0b .. n0b+47][0..255] -> LDS (coalesced float4)
    for (int idx = tid; idx < 48 * 64; idx += 96) {
        const int row = idx >> 6;
        const int kq  = (idx & 63) << 2;
        const float4 v = *(const float4*)(W2 + (size_t)(n0b + row) * HID + kq);
        *(float4*)(&bt[row * 260 + kq]) = v;
    }
    __syncthreads();

    const int n0 = n0b + wave * 16;
    const float* blds = bt + wave * 16 * 260;
    const int koff = (lane >> 4) << 1;           // 0 (lanes 0-15) or 2 (16-31)
    const int mrow = lane & 15;                  // A row within 16-tile / B col
    v8f acc0 = {}, acc1 = {}, acc2 = {}, acc3 = {};

    for (int kk = 0; kk < HID / 4; ++kk) {
        const int k0 = kk * 4 + koff;
        const v2f bf = *(const v2f*)(&blds[mrow * 260 + k0]);
        const v2f a0 = *(const v2f*)(hidden2_ws + (size_t)(mrow)      * HID + k0);
        const v2f a1 = *(const v2f*)(hidden2_ws + (size_t)(16 + mrow) * HID + k0);
        const v2f a2 = *(const v2f*)(hidden2_ws + (size_t)(32 + mrow) * HID + k0);
        const v2f a3 = *(const v2f*)(hidden2_ws + (size_t)(48 + mrow) * HID + k0);
        acc0 = __builtin_amdgcn_wmma_f32_16x16x4_f32(false, a0, false, bf, (short)0, acc0, false, false);
        acc1 = __builtin_amdgcn_wmma_f32_16x16x4_f32(false, a1, false, bf, (short)0, acc1, false, false);
        acc2 = __builtin_amdgcn_wmma_f32_16x16x4_f32(false, a2, false, bf, (short)0, acc2, false, false);
        acc3 = __builtin_amdgcn_wmma_f32_16x16x4_f32(false, a3, false, bf, (short)0, acc3, false, false);
    }

    // Epilogue: C layout — VGPR g, lanes 0-15 -> M=g, lanes 16-31 -> M=8+g.
    const int  half   = lane >> 4;
    const int  col    = n0 + (lane & 15);
    const float bias  = b2[col];
    const size_t mstr = (size_t)STEPS * VOCAB;
    float* obase = out + (size_t)step * VOCAB + col;
    #pragma unroll
    for (int g = 0; g < 8; ++g) {
        const int m = half * 8 + g;
        obase[(size_t)(m)      * mstr] = acc0[g] + bias;
        obase[(size_t)(m + 16) * mstr] = acc1[g] + bias;
        obase[(size_t)(m + 32) * mstr] = acc2[g] + bias;
        obase[(size_t)(m + 48) * mstr] = acc3[g] + bias;
    }
}

// ---------------------------------------------------------------------------
// Greedy argmax over V=30000 (first-occurrence tie-break, like jnp.argmax)
// ---------------------------------------------------------------------------
__global__ void argmax_kernel(const float* __restrict__ out, int* __restrict__ tok_ws, int step)
{
    const int b = blockIdx.x;
    const int j = threadIdx.x;
    const float* row = out + (size_t)b * STEPS * VOCAB + (size_t)step * VOCAB;
    __shared__ float bv[256];
    __shared__ int   bi[256];
    float best = -3.4e38f; int bidx = 0;
    for (int i = j; i < VOCAB; i += 256) {
        const float v = row[i];
        if (v > best) { best = v; bidx = i; }
    }
    bv[j] = best; bi[j] = bidx; __syncthreads();
    for (int off = 128; off > 0; off >>= 1) {
        if (j < off) {
            if (bv[j + off] > bv[j] || (bv[j + off] == bv[j] && bi[j + off] < bi[j])) {
                bv[j] = bv[j + off]; bi[j] = bi[j + off];
            }
        }
        __syncthreads();
    }
    if (j == 0) tok_ws[b] = bi[0];
}

// ---------------------------------------------------------------------------
extern "C" void kernel_launch(void* const* d_in, const int* in_sizes, int n_in,
                              void* d_out, int out_size, void* d_ws, size_t ws_size,
                              hipStream_t stream)
{
    (void)in_sizes; (void)n_in; (void)out_size; (void)ws_size;
    const int*   method_summary = (const int*)  d_in[0];
    const float* code_enc       = (const float*)d_in[1];
    const float* sbt_enc        = (const float*)d_in[2];
    const float* class_enc      = (const float*)d_in[3];
    const float* class_feat     = (const float*)d_in[4];
    const float* init_hidden    = (const float*)d_in[5];
    const float* emb            = (const float*)d_in[6];
    const float* W_ih           = (const float*)d_in[7];
    const float* W_hh           = (const float*)d_in[8];
    const float* b_ih           = (const float*)d_in[9];
    const float* b_hh           = (const float*)d_in[10];
    const float* Wq_uml         = (const float*)d_in[11];
    const float* bq_uml         = (const float*)d_in[12];
    const float* Wq_name        = (const float*)d_in[13];
    const float* bq_name        = (const float*)d_in[14];
    const float* Wv_uml         = (const float*)d_in[15];
    const float* bv_uml         = (const float*)d_in[16];
    const float* Wv_name        = (const float*)d_in[17];
    const float* bv_name        = (const float*)d_in[18];
    const float* W1             = (const float*)d_in[19];
    const float* b1             = (const float*)d_in[20];
    const float* W2             = (const float*)d_in[21];
    const float* b2             = (const float*)d_in[22];
    float* out = (float*)d_out;

    float* ws         = (float*)d_ws;
    float* h_ws       = ws;                       // 64*256
    float* hidden2_ws = ws + 16384;               // 64*256
    float* cq_ws      = ws + 32768;               // 64*2*256
    float* cv_ws      = ws + 65536;               // 64*2*256
    int*   tok_ws     = (int*)(ws + 98304);       // 64 ints

    setup_kernel<<<BATCH, HID, 0, stream>>>(class_enc, class_feat, init_hidden,
                                            method_summary,
                                            Wq_uml, bq_uml, Wq_name, bq_name,
                                            Wv_uml, bv_uml, Wv_name, bv_name,
                                            h_ws, cq_ws, cv_ws, tok_ws);

    for (int s = 0; s < STEPS; ++s) {
        gru_att_kernel<<<BATCH, HID, 0, stream>>>(tok_ws, emb, W_ih, W_hh, b_ih, b_hh,
                                                  code_enc, sbt_enc, cq_ws, cv_ws,
                                                  W1, b1, h_ws, hidden2_ws);
        logits_wmma_kernel<<<VOCAB / 48, 96, 0, stream>>>(hidden2_ws, W2, b2, out, s);
        if (s + 1 < STEPS)
            argmax_kernel<<<BATCH, HID, 0, stream>>>(out, tok_ws, s);
    }
}